// ForwardSumLoss_59820304499318
// MI455X (gfx1250) — compile-verified
//
#include <hip/hip_runtime.h>
#include <cstddef>

namespace {
constexpr int   kB    = 64;
constexpr int   kT    = 2000;          // T_OUT (query)
constexpr int   kK    = 400;           // T_IN (key)
constexpr int   kS    = 2 * kK + 1;    // 801 CTC states
constexpr int   kBlk2 = 832;           // 26 wave32s >= kS
constexpr float kNEG  = -1e30f;
constexpr float kBlank = -1.0f;
}

// ---------- CDNA5 async global->LDS helpers (ASYNCcnt path) ----------------
__device__ __forceinline__ void async_ld_f32(void* lds_dst, const void* gsrc) {
#if __has_builtin(__builtin_amdgcn_global_load_async_to_lds_b32)
  __builtin_amdgcn_global_load_async_to_lds_b32(
      (__attribute__((address_space(1))) int*)(void*)gsrc,
      (__attribute__((address_space(3))) int*)lds_dst,
      /*imm offset=*/0, /*cpol=*/0);
#else
  *(volatile float*)lds_dst = *(const float*)gsrc;  // synchronous fallback
#endif
}

__device__ __forceinline__ void wait_async0() {
#if __has_builtin(__builtin_amdgcn_s_wait_asynccnt)
  __builtin_amdgcn_s_wait_asynccnt(0);
#elif __has_builtin(__builtin_amdgcn_global_load_async_to_lds_b32)
  asm volatile("s_wait_asynccnt 0" ::: "memory");
#endif
}

// ---------------------------------------------------------------------------
// Phase 1: masked log-softmax normalizer logZ[b*T+t].
// One wave32 per (b,t) row. Classes 1..400 streamed as float4 (B128) loads;
// blank (class 0, value -1.0) folded in analytically.
// ---------------------------------------------------------------------------
__global__ __launch_bounds__(256) void logz_kernel(
    const float* __restrict__ attn, const int* __restrict__ in_lens,
    const int* __restrict__ out_lens, float* __restrict__ logZ) {
  const int gid  = blockIdx.x * blockDim.x + threadIdx.x;
  const int row  = gid >> 5;             // (b,t) row per wave32
  const int lane = threadIdx.x & 31;
  if (row >= kB * kT) return;
  const int b = row / kT;
  const int t = row - b * kT;
  if (t >= out_lens[b]) return;          // alpha frozen there: logZ never read
  const int in_len = in_lens[b];
  const float* __restrict__ r = attn + ((size_t)b * kT + t) * kK;

  float x[16];
  float m = kBlank;                      // blank always valid, folded into max
#pragma unroll
  for (int k = 0; k < 4; ++k) {
    const int base = (lane << 2) + (k << 7);   // float index into row, 0..508
    float4 v4 = make_float4(kNEG, kNEG, kNEG, kNEG);
    if (base + 3 < kK) v4 = *(const float4*)(r + base);  // fully in-row quads
    const float vv[4] = {v4.x, v4.y, v4.z, v4.w};
#pragma unroll
    for (int e = 0; e < 4; ++e) {
      const int c = base + e + 1;        // class index 1..kK
      const float v = (c <= in_len) ? vv[e] : kNEG;
      x[k * 4 + e] = v;
      m = fmaxf(m, v);
    }
  }
#pragma unroll
  for (int off = 16; off > 0; off >>= 1) m = fmaxf(m, __shfl_xor(m, off, 32));
  float s = 0.f;
#pragma unroll
  for (int k = 0; k < 16; ++k) s += __expf(x[k] - m);   // exp(-1e30)==0 masks
#pragma unroll
  for (int off = 16; off > 0; off >>= 1) s += __shfl_xor(s, off, 32);
  if (lane == 0) logZ[row] = m + __logf(s + __expf(kBlank - m));
}

// ---------------------------------------------------------------------------
// Phase 2: CTC alpha recursion. One block per sample b; thread s owns state s.
// alpha[s] carried in a register; neighbors via double-buffered LDS
// (1 barrier/step). Emission for t+1 prefetched via async global->LDS; the
// wave-local s_wait_asynccnt sits AFTER the barrier so the DMA overlaps both
// the alpha LDS store and the barrier latency.
// ---------------------------------------------------------------------------
__global__ __launch_bounds__(kBlk2) void ctc_alpha_kernel(
    const float* __restrict__ attn, const int* __restrict__ in_lens,
    const int* __restrict__ out_lens, const float* __restrict__ logZ,
    float* __restrict__ lossBuf) {
  __shared__ float A[2][kS + 2];     // alpha, 2-slot NEG pad at front
  __shared__ float EM[2][kBlk2];     // async-prefetched emissions (odd states)

  const int b = blockIdx.x;
  const int s = threadIdx.x;
  const int in_len  = in_lens[b];
  const int tEnd    = out_lens[b];   // alpha frozen for t >= out_len: stop
  const bool active = (s < kS);
  const bool odd    = (s & 1) != 0;
  const int  j      = (s + 1) >> 1;                 // class emitted by state s
  const bool tokvalid = !odd || (j <= in_len);
  const bool loads    = active && odd;              // j in 1..kK for odd s<=799

  const float* __restrict__ row0 = attn + (size_t)b * kT * kK;
  const float* __restrict__ gsrc0 = row0 + (j - 1); // this thread's class col
  const float* __restrict__ lz = logZ + (size_t)b * kT;

  if (s < 2) { A[0][s] = kNEG; A[1][s] = kNEG; }    // pads (only buffer edges)

  // ---- t = 0: prefetch emissions, init alpha ----
  if (loads) async_ld_f32(&EM[0][s], gsrc0);
  wait_async0();                    // own-lane data: no barrier needed to read
  const float z0 = lz[0];
  float em0;
  if (!active)       em0 = kNEG;
  else if (!odd)     em0 = kBlank - z0;
  else if (tokvalid) em0 = EM[0][s] - z0;
  else               em0 = kNEG;
  float a0 = (active && s <= 1) ? em0 : kNEG;       // register-carried alpha[s]
  if (active) A[0][s + 2] = a0;

  // prefetch t=1 (consumed after the barrier; wave-local wait at loop top)
  if (tEnd > 1 && loads) async_ld_f32(&EM[1][s], gsrc0 + (size_t)kK);
  float z_next = (tEnd > 1) ? lz[1] : 0.f;
  __syncthreads();

  int acur = 0;
  for (int t = 1; t < tEnd; ++t) {
    wait_async0();                 // completes the prefetch issued last iter
    const float z = z_next;
    float em;
    if (!active)       em = kNEG;
    else if (!odd)     em = kBlank - z;
    else if (tokvalid) em = EM[t & 1][s] - z;
    else               em = kNEG;

    if (active) {
      const float a1 = A[acur][s + 1];
      const float a2 = (odd && s >= 3) ? A[acur][s] : kNEG;
      const float m  = fmaxf(fmaxf(a0, a1), a2);
      const float sm = __expf(a0 - m) + __expf(a1 - m) + __expf(a2 - m);
      a0 = em + (m + __logf(sm));
    }

    // overlap: issue prefetch of t+1 before the store/barrier
    if (t + 1 < tEnd) {
      if (loads) async_ld_f32(&EM[(t + 1) & 1][s], gsrc0 + (size_t)(t + 1) * kK);
      z_next = lz[t + 1];
    }

    if (active) A[acur ^ 1][s + 2] = a0;
    __syncthreads();
    acur ^= 1;
  }

  if (s == 0) {
    const int last = 2 * in_len;                     // final blank state
    const float aS   = A[acur][last + 2];
    const float aSm1 = A[acur][last + 1];
    const float m  = fmaxf(aS, aSm1);
    const float ll = m + __logf(__expf(aS - m) + __expf(aSm1 - m));
    float loss = -ll / (float)in_len;
    const bool ok = isfinite(loss) && (ll > kNEG * 0.5f);
    lossBuf[b] = ok ? loss : 0.0f;
  }
}

// ---------------------------------------------------------------------------
// Phase 3: deterministic fixed-order mean over B samples.
// ---------------------------------------------------------------------------
__global__ void finalize_kernel(const float* __restrict__ lossBuf,
                                float* __restrict__ out) {
  if (threadIdx.x == 0 && blockIdx.x == 0) {
    float s = 0.f;
    for (int i = 0; i < kB; ++i) s += lossBuf[i];
    out[0] = s / (float)kB;
  }
}

// ---------------------------------------------------------------------------
extern "C" void kernel_launch(void* const* d_in, const int* in_sizes, int n_in,
                              void* d_out, int out_size, void* d_ws,
                              size_t ws_size, hipStream_t stream) {
  const float* attn     = (const float*)d_in[0];  // (B,1,T,K) f32
  const int*   in_lens  = (const int*)d_in[1];    // (B,) i32
  const int*   out_lens = (const int*)d_in[2];    // (B,) i32
  float*       out      = (float*)d_out;          // scalar mean loss

  float* logZ    = (float*)d_ws;                  // B*T floats
  float* lossBuf = logZ + (size_t)kB * kT;        // B floats

  // Phase 1: one wave32 per (b,t) row -> 128000 waves, 256-thr blocks
  const int rows = kB * kT;
  const int blocks1 = (rows * 32 + 255) / 256;
  logz_kernel<<<blocks1, 256, 0, stream>>>(attn, in_lens, out_lens, logZ);

  // Phase 2: one 832-thread workgroup per sample (fills 64 WGPs)
  ctc_alpha_kernel<<<kB, kBlk2, 0, stream>>>(attn, in_lens, out_lens, logZ,
                                             lossBuf);

  // Phase 3: scalar reduction
  finalize_kernel<<<1, 32, 0, stream>>>(lossBuf, out);
}